// RevGNN_21440476741726
// MI455X (gfx1250) — compile-verified
//
#include <hip/hip_runtime.h>
#include <math.h>

// Problem constants (match reference)
#define NN   100000
#define EE   1600000
#define INC  128
#define HC   128
#define OUTC 47
#define LL   3
#define GG   2
#define CC   64
#define EPSV 1e-5f

typedef float v2f __attribute__((ext_vector_type(2)));
typedef float v8f __attribute__((ext_vector_type(8)));

__device__ __forceinline__ v8f wmma_k4(v2f a, v2f b, v8f c) {
  // V_WMMA_F32_16X16X4_F32 : D = A(16x4) * B(4x16) + C(16x16), fp32 throughout
  return __builtin_amdgcn_wmma_f32_16x16x4_f32(false, a, false, b, (short)0, c,
                                               false, false);
}

// ---------------------------------------------------------------- utilities
__global__ void zero_kernel(float* __restrict__ p, int n) {
  int i = blockIdx.x * blockDim.x + threadIdx.x;
  int stride = gridDim.x * blockDim.x;
  for (; i < n; i += stride) p[i] = 0.0f;
}

__global__ void deg_count_kernel(const int* __restrict__ dst,
                                 float* __restrict__ deg) {
  int e = blockIdx.x * 256 + threadIdx.x;
  if (e < EE) atomicAdd(&deg[dst[e]], 1.0f);
}

__global__ void deg_finalize_kernel(float* __restrict__ deg) {
  int i = blockIdx.x * 256 + threadIdx.x;
  if (i < NN) deg[i] = 1.0f / fmaxf(deg[i], 1.0f);
}

// ------------------------------------------------- encoder: h = x@We + be
// grid = N/16 blocks, 256 threads = 8 waves, wave w -> cols [16w,16w+16)
__global__ __launch_bounds__(256) void encoder_gemm_kernel(
    const float* __restrict__ X, const float* __restrict__ W,
    const float* __restrict__ bias, float* __restrict__ Hout) {
  const int wave = threadIdx.x >> 5;
  const int lane = threadIdx.x & 31;
  const int m0 = blockIdx.x * 16;
  const int n0 = wave * 16;
  const int hi  = lane >> 4;    // 0: K{0,1}, 1: K{2,3}
  const int l16 = lane & 15;
  const int am = m0 + l16;      // A row held by this lane
  const int bn = n0 + l16;      // B col held by this lane
  v8f acc = {};
  for (int k0 = 0; k0 < INC; k0 += 4) {
    const int ka = k0 + 2 * hi;
    v2f a, b;
    a.x = X[am * INC + ka];
    a.y = X[am * INC + ka + 1];
    b.x = W[ka * HC + bn];
    b.y = W[(ka + 1) * HC + bn];
    acc = wmma_k4(a, b, acc);
  }
  const float bb = bias[bn];
#pragma unroll
  for (int r = 0; r < 8; ++r) {
    const int m = m0 + r + 8 * hi;
    Hout[m * HC + bn] = acc[r] + bb;
  }
}

// -------------------------------------- z = relu(layernorm(y)) over 64 ch
// one wave per node, 2 channels/lane
__global__ __launch_bounds__(256) void ln_relu64_kernel(
    const float* __restrict__ Y, const float* __restrict__ gamma,
    const float* __restrict__ beta, float* __restrict__ Z) {
  const int node = blockIdx.x * 8 + (threadIdx.x >> 5);
  const int lane = threadIdx.x & 31;
  if (node >= NN) return;
  const float* row = Y + (size_t)node * HC;  // caller passes slice base
  float2 v = *(const float2*)(row + 2 * lane);
  float s  = v.x + v.y;
  float sq = v.x * v.x + v.y * v.y;
#pragma unroll
  for (int off = 16; off > 0; off >>= 1) {
    s  += __shfl_xor(s, off, 32);
    sq += __shfl_xor(sq, off, 32);
  }
  const float mean = s * (1.0f / 64.0f);
  const float var  = sq * (1.0f / 64.0f) - mean * mean;
  const float rstd = rsqrtf(var + EPSV);
  float2 g2 = *(const float2*)(gamma + 2 * lane);
  float2 b2 = *(const float2*)(beta + 2 * lane);
  float2 o;
  o.x = fmaxf(0.0f, (v.x - mean) * rstd * g2.x + b2.x);
  o.y = fmaxf(0.0f, (v.y - mean) * rstd * g2.y + b2.y);
  *(float2*)(Z + (size_t)node * CC + 2 * lane) = o;
}

// -------------------------------------- agg[dst] += z[src] (scatter add)
// thread-per-(edge, channel): 64 consecutive threads = one edge row (256B)
__global__ __launch_bounds__(256) void aggregate_kernel(
    const float* __restrict__ Z, const int* __restrict__ src,
    const int* __restrict__ dst, float* __restrict__ AGG) {
  const int t = blockIdx.x * 256 + threadIdx.x;  // < EE*64 = 102.4M
  const int e  = t >> 6;
  const int ch = t & 63;
  if (e < EE) {
    const int s = src[e];
    const int d = dst[e];
    atomicAdd(&AGG[(size_t)d * CC + ch], Z[(size_t)s * CC + ch]);
  }
}

// ---- y_out = x_g + (agg .* deginv) @ Wl + bl + z @ Wr   (fused K=128 GEMM)
// grid = N/16 blocks, 128 threads = 4 waves covering the 64 output cols
__global__ __launch_bounds__(128) void group_update_kernel(
    const float* __restrict__ AGG, const float* __restrict__ DINV,
    const float* __restrict__ Z, const float* __restrict__ Wl,
    const float* __restrict__ bl, const float* __restrict__ Wr,
    const float* __restrict__ XG, float* __restrict__ Yout) {
  const int wave = threadIdx.x >> 5;
  const int lane = threadIdx.x & 31;
  const int m0 = blockIdx.x * 16;
  const int n0 = wave * 16;
  const int hi  = lane >> 4;
  const int l16 = lane & 15;
  const int am = m0 + l16;
  const int bn = n0 + l16;
  const float dinv = DINV[am];
  v8f acc = {};
  // K = 0..63 : mean-aggregated neighbors x Wl
  for (int k0 = 0; k0 < CC; k0 += 4) {
    const int ka = k0 + 2 * hi;
    v2f a, b;
    a.x = AGG[(size_t)am * CC + ka] * dinv;
    a.y = AGG[(size_t)am * CC + ka + 1] * dinv;
    b.x = Wl[ka * CC + bn];
    b.y = Wl[(ka + 1) * CC + bn];
    acc = wmma_k4(a, b, acc);
  }
  // K = 64..127 : self features x Wr
  for (int k0 = 0; k0 < CC; k0 += 4) {
    const int ka = k0 + 2 * hi;
    v2f a, b;
    a.x = Z[(size_t)am * CC + ka];
    a.y = Z[(size_t)am * CC + ka + 1];
    b.x = Wr[ka * CC + bn];
    b.y = Wr[(ka + 1) * CC + bn];
    acc = wmma_k4(a, b, acc);
  }
  const float bb = bl[bn];
#pragma unroll
  for (int r = 0; r < 8; ++r) {
    const int m = m0 + r + 8 * hi;
    Yout[m * HC + bn] = XG[m * HC + bn] + bb + acc[r];  // reversible residual
  }
}

// ---------------------------- final LN+ReLU over all 128 ch, wave per node
__global__ __launch_bounds__(256) void ln_relu128_kernel(
    const float* __restrict__ H, const float* __restrict__ gamma,
    const float* __restrict__ beta, float* __restrict__ Out) {
  const int node = blockIdx.x * 8 + (threadIdx.x >> 5);
  const int lane = threadIdx.x & 31;
  if (node >= NN) return;
  const float* row = H + (size_t)node * HC;
  float4 v = *(const float4*)(row + 4 * lane);
  float s  = v.x + v.y + v.z + v.w;
  float sq = v.x * v.x + v.y * v.y + v.z * v.z + v.w * v.w;
#pragma unroll
  for (int off = 16; off > 0; off >>= 1) {
    s  += __shfl_xor(s, off, 32);
    sq += __shfl_xor(sq, off, 32);
  }
  const float mean = s * (1.0f / 128.0f);
  const float var  = sq * (1.0f / 128.0f) - mean * mean;
  const float rstd = rsqrtf(var + EPSV);
  float4 g4 = *(const float4*)(gamma + 4 * lane);
  float4 b4 = *(const float4*)(beta + 4 * lane);
  float4 o;
  o.x = fmaxf(0.0f, (v.x - mean) * rstd * g4.x + b4.x);
  o.y = fmaxf(0.0f, (v.y - mean) * rstd * g4.y + b4.y);
  o.z = fmaxf(0.0f, (v.z - mean) * rstd * g4.z + b4.z);
  o.w = fmaxf(0.0f, (v.w - mean) * rstd * g4.w + b4.w);
  *(float4*)(Out + (size_t)node * HC + 4 * lane) = o;
}

// --------------------------- logits = hn @ Wp + bp  (K=128, 47 cols padded)
// grid = N/16 blocks, 96 threads = 3 waves (cols 0..47; col 47 masked).
// Loads are clamped (not predicated) so EXEC is all-ones at every WMMA.
__global__ __launch_bounds__(96) void classifier_gemm_kernel(
    const float* __restrict__ Hn, const float* __restrict__ Wp,
    const float* __restrict__ bp, float* __restrict__ Logits) {
  const int wave = threadIdx.x >> 5;
  const int lane = threadIdx.x & 31;
  const int m0 = blockIdx.x * 16;
  const int n0 = wave * 16;
  const int hi  = lane >> 4;
  const int l16 = lane & 15;
  const int am = m0 + l16;
  const int bn = n0 + l16;
  const bool nvalid = bn < OUTC;
  const int bnc = nvalid ? bn : (OUTC - 1);  // clamp: keep loads unconditional
  v8f acc = {};
  for (int k0 = 0; k0 < HC; k0 += 4) {
    const int ka = k0 + 2 * hi;
    v2f a, b;
    a.x = Hn[(size_t)am * HC + ka];
    a.y = Hn[(size_t)am * HC + ka + 1];
    const float w0 = Wp[ka * OUTC + bnc];
    const float w1 = Wp[(ka + 1) * OUTC + bnc];
    b.x = nvalid ? w0 : 0.0f;
    b.y = nvalid ? w1 : 0.0f;
    acc = wmma_k4(a, b, acc);
  }
  const float bb = bp[bnc];
#pragma unroll
  for (int r = 0; r < 8; ++r) {
    const int m = m0 + r + 8 * hi;
    if (nvalid) Logits[(size_t)m * OUTC + bn] = acc[r] + bb;
  }
}

// --------------------------------------- row-wise log_softmax over 47 cols
__global__ __launch_bounds__(256) void log_softmax_kernel(
    const float* __restrict__ Logits, float* __restrict__ Out) {
  const int node = blockIdx.x * 8 + (threadIdx.x >> 5);
  const int lane = threadIdx.x & 31;
  if (node >= NN) return;
  const float* row = Logits + (size_t)node * OUTC;
  const float v0 = row[lane];  // lane < 32 <= 46 always valid
  const bool has1 = (lane + 32) < OUTC;
  const float v1 = has1 ? row[lane + 32] : -3.4e38f;
  float mx = fmaxf(v0, v1);
#pragma unroll
  for (int off = 16; off > 0; off >>= 1) mx = fmaxf(mx, __shfl_xor(mx, off, 32));
  float e = __expf(v0 - mx) + (has1 ? __expf(v1 - mx) : 0.0f);
#pragma unroll
  for (int off = 16; off > 0; off >>= 1) e += __shfl_xor(e, off, 32);
  const float lse = mx + __logf(e);
  if (lane < OUTC) Out[(size_t)node * OUTC + lane] = v0 - lse;
  if (has1) Out[(size_t)node * OUTC + lane + 32] = v1 - lse;
}

// ---------------------------------------------------------------- launcher
extern "C" void kernel_launch(void* const* d_in, const int* in_sizes, int n_in,
                              void* d_out, int out_size, void* d_ws,
                              size_t ws_size, hipStream_t stream) {
  const float* x     = (const float*)d_in[0];
  const float* We    = (const float*)d_in[1];
  const float* be    = (const float*)d_in[2];
  const float* ln_g  = (const float*)d_in[3];
  const float* ln_b  = (const float*)d_in[4];
  const float* Wl    = (const float*)d_in[5];
  const float* bl    = (const float*)d_in[6];
  const float* Wr    = (const float*)d_in[7];
  const float* lnf_g = (const float*)d_in[8];
  const float* lnf_b = (const float*)d_in[9];
  const float* Wp    = (const float*)d_in[10];
  const float* bp    = (const float*)d_in[11];
  const int*   ei    = (const int*)d_in[12];
  const int* srcI = ei;
  const int* dstI = ei + EE;

  float* ws = (float*)d_ws;
  float* deginv = ws;  ws += NN;
  float* hA     = ws;  ws += (size_t)NN * HC;
  float* hB     = ws;  ws += (size_t)NN * HC;
  float* zbuf   = ws;  ws += (size_t)NN * CC;
  float* agg    = ws;  ws += (size_t)NN * CC;
  float* logits = ws;  ws += (size_t)NN * OUTC;

  const int zgrid = 2048;

  // in-degree -> 1/max(deg,1)
  zero_kernel<<<zgrid, 256, 0, stream>>>(deginv, NN);
  deg_count_kernel<<<EE / 256, 256, 0, stream>>>(dstI, deginv);
  deg_finalize_kernel<<<(NN + 255) / 256, 256, 0, stream>>>(deginv);

  // encoder
  encoder_gemm_kernel<<<NN / 16, 256, 0, stream>>>(x, We, be, hA);

  float* cur = hA;
  float* nxt = hB;
  for (int l = 0; l < LL; ++l) {
    // ---- group 0: y = cur[:, 64:128]
    ln_relu64_kernel<<<NN / 8, 256, 0, stream>>>(
        cur + CC, ln_g + (size_t)(l * GG + 0) * CC,
        ln_b + (size_t)(l * GG + 0) * CC, zbuf);
    zero_kernel<<<zgrid, 256, 0, stream>>>(agg, NN * CC);
    aggregate_kernel<<<(EE * 64) / 256, 256, 0, stream>>>(zbuf, srcI, dstI, agg);
    group_update_kernel<<<NN / 16, 128, 0, stream>>>(
        agg, deginv, zbuf, Wl + (size_t)(l * GG + 0) * CC * CC,
        bl + (size_t)(l * GG + 0) * CC, Wr + (size_t)(l * GG + 0) * CC * CC,
        cur + 0, nxt + 0);
    // ---- group 1: y = nxt[:, 0:64] (the group-0 output)
    ln_relu64_kernel<<<NN / 8, 256, 0, stream>>>(
        nxt + 0, ln_g + (size_t)(l * GG + 1) * CC,
        ln_b + (size_t)(l * GG + 1) * CC, zbuf);
    zero_kernel<<<zgrid, 256, 0, stream>>>(agg, NN * CC);
    aggregate_kernel<<<(EE * 64) / 256, 256, 0, stream>>>(zbuf, srcI, dstI, agg);
    group_update_kernel<<<NN / 16, 128, 0, stream>>>(
        agg, deginv, zbuf, Wl + (size_t)(l * GG + 1) * CC * CC,
        bl + (size_t)(l * GG + 1) * CC, Wr + (size_t)(l * GG + 1) * CC * CC,
        cur + CC, nxt + CC);
    float* t = cur; cur = nxt; nxt = t;
  }

  // head: LN+ReLU -> classifier GEMM -> log_softmax
  ln_relu128_kernel<<<NN / 8, 256, 0, stream>>>(cur, lnf_g, lnf_b, nxt);
  classifier_gemm_kernel<<<NN / 16, 96, 0, stream>>>(nxt, Wp, bp, logits);
  log_softmax_kernel<<<NN / 8, 256, 0, stream>>>(logits, (float*)d_out);
}